// ColBertPairwiseDistillKLLoss_9740985828096
// MI455X (gfx1250) — compile-verified
//
#include <hip/hip_runtime.h>
#include <math.h>
#include <stdint.h>

typedef __attribute__((ext_vector_type(2))) float v2f;
typedef __attribute__((ext_vector_type(8))) float v8f;

#define B_SZ   64
#define N_Q    32
#define S_D    1024
#define D_DIM  128
#define PITCH  132           // 128 + 4 pad: row stride mod 64 banks = 4 -> conflict-free
#define CHUNK_S 64
#define NCHUNK (S_D / CHUNK_S)

// ---------------------------------------------------------------------------
// Kernel 1: MaxSim scores. One workgroup (8 wave32) per (b,c) pair.
// wave w: nTile = w&1 (16-row tile of N=32), sTile = w>>1 (16-col tile of the
// 64-col S chunk). A fragments (Q tile, all K=128) hoisted into 64 VGPRs once;
// D chunks streamed into LDS with GLOBAL_LOAD_ASYNC_TO_LDS_B128 (ASYNCcnt).
// Per chunk: all 32 B fragments batch-loaded (sched_barrier keeps the DS loads
// pipelined ahead of the math), then 32 V_WMMA_F32_16X16X4_F32 in two
// independent accumulation chains.
// ---------------------------------------------------------------------------
__launch_bounds__(256)
__global__ void colbert_maxsim_kernel(const float* __restrict__ Q,
                                      const float* __restrict__ Dm,
                                      float* __restrict__ scores) {
  __shared__ float Ds[CHUNK_S * PITCH];   // 33,792 B: Q staging, D chunks, reduction
  __shared__ float rmx[N_Q];

  const int pair = blockIdx.x;
  const int b = pair >> 6;
  const int c = pair & 63;
  const int tid  = threadIdx.x;
  const int lane = tid & 31;
  const int wave = tid >> 5;
  const int nTile = wave & 1;
  const int sTile = wave >> 1;

  const float*  Qg  = Q  + (size_t)b * N_Q * D_DIM;
  const float*  Dg  = Dm + (size_t)c * S_D * D_DIM;
  const float4* Qg4 = (const float4*)Qg;

  // ---- Stage Q_b (32 x 128 f32) into the shared buffer (transient) --------
  #pragma unroll
  for (int i = 0; i < 4; ++i) {
    int idx = tid + i * 256;     // 1024 float4
    int row = idx >> 5;          // 32 float4 per row
    int c4  = idx & 31;
    float4 v = Qg4[idx];
    *(float4*)&Ds[row * PITCH + c4 * 4] = v;
  }
  __syncthreads();

  // ---- Hoist this wave's A fragments into registers -----------------------
  // WMMA fp32 16x16x4 A layout: lanes 0-15 hold M=lane with k-pair {0,1},
  // lanes 16-31 hold k-pair {2,3}.
  const int m  = lane & 15;
  const int kh = (lane >> 4) << 1;    // 0 or 2
  v2f afrag[32];
  {
    const float* Arow = &Ds[(nTile * 16 + m) * PITCH];
    #pragma unroll
    for (int j = 0; j < 32; ++j) {
      afrag[j] = *(const v2f*)(Arow + 4 * j + kh);   // 8B-aligned LDS b64
    }
  }
  __syncthreads();   // all waves done reading Q before D chunks overwrite it

  const float* Brow = &Ds[(sTile * 16 + m) * PITCH];

  float runMax[8];
  #pragma unroll
  for (int v = 0; v < 8; ++v) runMax[v] = -INFINITY;

  for (int chunk = 0; chunk < NCHUNK; ++chunk) {
    __syncthreads();   // previous chunk's LDS reads complete (waits precede use)

    // ---- Async stage 64 x 128 chunk of D_c straight into LDS --------------
    // 2048 b128 transfers: 8 per thread, tracked by ASYNCcnt (no VGPR bounce).
    #pragma unroll
    for (int i = 0; i < 8; ++i) {
      int idx = tid + i * 256;
      int row = idx >> 5;
      int c4  = idx & 31;
      const float* gsrc = Dg + (size_t)(chunk * CHUNK_S + row) * D_DIM + c4 * 4;
      uint32_t ldst = (uint32_t)(uintptr_t)&Ds[row * PITCH + c4 * 4];
      asm volatile("global_load_async_to_lds_b128 %0, %1, off"
                   :: "v"(ldst), "v"((uint64_t)(uintptr_t)gsrc)
                   : "memory");
    }
    asm volatile("s_wait_asynccnt 0x0" ::: "memory");
    __syncthreads();

    // L2 prefetch of the next chunk (global_prefetch_b8).
    if (chunk + 1 < NCHUNK) {
      __builtin_prefetch(Dg + (size_t)(chunk + 1) * CHUNK_S * D_DIM + tid * 32, 0, 1);
    }

    // ---- Batch-load all 32 B fragments; keep DS loads ahead of the math ---
    v2f bfrag[32];
    #pragma unroll
    for (int j = 0; j < 32; ++j) {
      bfrag[j] = *(const v2f*)(Brow + 4 * j + kh);   // 8B-aligned LDS b64
    }
    __builtin_amdgcn_sched_barrier(0);   // loads may not sink below this point

    // ---- 32 fp32 WMMAs over K=128 in two independent chains ---------------
    v8f acc0 = {};
    v8f acc1 = {};
    #pragma unroll
    for (int j = 0; j < 32; j += 2) {
      acc0 = __builtin_amdgcn_wmma_f32_16x16x4_f32(
          false, afrag[j],     false, bfrag[j],
          (short)0, acc0, false, false);
      acc1 = __builtin_amdgcn_wmma_f32_16x16x4_f32(
          false, afrag[j + 1], false, bfrag[j + 1],
          (short)0, acc1, false, false);
    }
    // Running max over S: element v -> n = v (lanes 0-15) or v+8 (lanes
    // 16-31), s = chunk base + sTile*16 + (lane&15).
    #pragma unroll
    for (int v = 0; v < 8; ++v) {
      runMax[v] = fmaxf(runMax[v], acc0[v] + acc1[v]);
    }
  }

  // ---- Reduction: per-row max over s-tiles/lanes, then sum over n ---------
  __syncthreads();
  float* red = Ds;   // reuse chunk buffer
  #pragma unroll
  for (int v = 0; v < 8; ++v) red[(wave * 32 + lane) * 8 + v] = runMax[v];
  __syncthreads();

  if (tid < 32) {
    int n  = tid;
    int nt = n >> 4;
    int r  = n & 15;
    int v  = r & 7;
    int lh = (r & 8) ? 16 : 0;
    float mx = -INFINITY;
    for (int sT = 0; sT < 4; ++sT) {
      int w = (sT << 1) | nt;
      for (int l = lh; l < lh + 16; ++l)
        mx = fmaxf(mx, red[(w * 32 + l) * 8 + v]);
    }
    rmx[n] = mx;
  }
  __syncthreads();
  if (tid == 0) {
    float s = 0.f;
    #pragma unroll
    for (int n = 0; n < N_Q; ++n) s += rmx[n];
    scores[pair] = s;
  }
}

// ---------------------------------------------------------------------------
// Kernel 2: scalar loss from the two 64x64 score matrices.
// contrastive = mean_b softplus(hardest_neg - pos)
// kl = batchmean KL(teacher || student log-softmax), T=1, ALPHA=0.5
// ---------------------------------------------------------------------------
__launch_bounds__(64)
__global__ void colbert_loss_kernel(const float* __restrict__ S,
                                    const float* __restrict__ T,
                                    float* __restrict__ out) {
  __shared__ float lbuf[B_SZ];
  __shared__ float kbuf[B_SZ];
  const int b = threadIdx.x;
  const float* srow = S + b * B_SZ;
  const float* trow = T + b * B_SZ;

  float pos = srow[b];
  float neg = -INFINITY, smax = -INFINITY, tmax = -INFINITY;
  for (int c = 0; c < B_SZ; ++c) {
    float sv = srow[c], tv = trow[c];
    if (c != b) neg = fmaxf(neg, sv);
    smax = fmaxf(smax, sv);
    tmax = fmaxf(tmax, tv);
  }
  float ssum = 0.f, tsum = 0.f;
  for (int c = 0; c < B_SZ; ++c) {
    ssum += expf(srow[c] - smax);
    tsum += expf(trow[c] - tmax);
  }
  float lse  = smax + logf(ssum);
  float tlse = tmax + logf(tsum);
  float kl = 0.f;
  for (int c = 0; c < B_SZ; ++c) {
    float tlogp = trow[c] - tlse;              // log teacher prob
    float slogp = srow[c] - lse;               // student log prob
    kl += expf(tlogp) * (tlogp - slogp);
  }
  float x  = neg - pos;
  float sp = (x > 0.f) ? (x + log1pf(expf(-x))) : log1pf(expf(x));
  lbuf[b] = sp;
  kbuf[b] = kl;
  __syncthreads();
  if (b == 0) {
    float ls = 0.f, ks = 0.f;
    for (int i = 0; i < B_SZ; ++i) { ls += lbuf[i]; ks += kbuf[i]; }
    // TEMPERATURE = 1 -> T^2 = 1; ALPHA = 0.5
    out[0] = ls / (float)B_SZ + 0.5f * (ks / (float)B_SZ);
  }
}

extern "C" void kernel_launch(void* const* d_in, const int* in_sizes, int n_in,
                              void* d_out, int out_size, void* d_ws, size_t ws_size,
                              hipStream_t stream) {
  const float* q  = (const float*)d_in[0];   // [64,32,128]
  const float* d  = (const float*)d_in[1];   // [64,1024,128]
  const float* tq = (const float*)d_in[2];   // [64,32,128]
  const float* td = (const float*)d_in[3];   // [64,1024,128]

  float* sS = (float*)d_ws;                  // student scores [64,64]
  float* sT = sS + B_SZ * B_SZ;              // teacher scores [64,64]

  dim3 grid(B_SZ * B_SZ);
  dim3 block(256);
  colbert_maxsim_kernel<<<grid, block, 0, stream>>>(q, d, sS);
  colbert_maxsim_kernel<<<grid, block, 0, stream>>>(tq, td, sT);
  colbert_loss_kernel<<<1, B_SZ, 0, stream>>>(sS, sT, (float*)d_out);
}